// SpeAttention_24412594111032
// MI455X (gfx1250) — compile-verified
//
#include <hip/hip_runtime.h>
#include <math.h>

typedef __attribute__((ext_vector_type(2))) float v2f;
typedef __attribute__((ext_vector_type(4))) float fv4;
typedef __attribute__((ext_vector_type(8))) float v8f;

#define CH   32
#define LSEQ 128
#define NST  8

__device__ __forceinline__ float silu_f(float x) { return x / (1.0f + expf(-x)); }

// ---------------------------------------------------------------------------
// Kernel 1: streaming mean+max reduction over each [512x512] slice.
// 1024 blocks (one per (b,c)), 1024 threads, b128 non-temporal loads.
// ws[0..1023] = mean, ws[1024..2047] = max.
// ---------------------------------------------------------------------------
__global__ __launch_bounds__(1024) void k_reduce(const float* __restrict__ in,
                                                 float* __restrict__ ws) {
  const int s = blockIdx.x;                       // slice id 0..1023
  const fv4* p4 = (const fv4*)(in + (size_t)s * (512 * 512));
  float sum = 0.0f;
  float mx  = -3.402823466e38f;
  #pragma unroll 8
  for (int i = threadIdx.x; i < 65536; i += 1024) {
    fv4 v = __builtin_nontemporal_load(p4 + i);   // global_load_b128 th:NT
    sum += (v.x + v.y) + (v.z + v.w);
    mx = fmaxf(mx, fmaxf(fmaxf(v.x, v.y), fmaxf(v.z, v.w)));
  }
  // wave32 reduce
  #pragma unroll
  for (int o = 16; o > 0; o >>= 1) {
    sum += __shfl_xor(sum, o, 32);
    mx = fmaxf(mx, __shfl_xor(mx, o, 32));
  }
  __shared__ float ssum[32];
  __shared__ float smax[32];
  const int wave = threadIdx.x >> 5, lane = threadIdx.x & 31;
  if (lane == 0) { ssum[wave] = sum; smax[wave] = mx; }
  __syncthreads();
  if (wave == 0) {
    float s2 = ssum[lane];
    float m2 = smax[lane];
    #pragma unroll
    for (int o = 16; o > 0; o >>= 1) {
      s2 += __shfl_xor(s2, o, 32);
      m2 = fmaxf(m2, __shfl_xor(m2, o, 32));
    }
    if (lane == 0) {
      ws[s]        = s2 * (1.0f / 262144.0f);
      ws[1024 + s] = m2;
    }
  }
}

// ---------------------------------------------------------------------------
// Kernel 2: fused lin1 + LN + bimamba-v2 + LN + (out_proj*lin2) + tanh.
// grid = 8 (batch), block = 256 (8 waves).
//   sA : h (phase 1-2), then dt (phase 3)   [128x32]
//   sX : x (conv input)                      [128x32]
//   sXC: conv output (per branch)            [128x32]
//   sXD: xdbl (dt_raw|B|C)                   [128x18]
//   z and ysum live in global workspace (write-once/read-once).
// ---------------------------------------------------------------------------
__global__ __launch_bounds__(256) void k_mamba(
    const float* __restrict__ ws,   // avg[1024], mx[1024]
    float* __restrict__ zws,        // [8][128][32]
    float* __restrict__ yws,        // [8][128][32]
    const float* __restrict__ w1,   const float* __restrict__ b1,
    const float* __restrict__ g1,   const float* __restrict__ bb1,
    const float* __restrict__ win,  // [64][32]
    const float* __restrict__ cwf,  const float* __restrict__ cbf,
    const float* __restrict__ cwb,  const float* __restrict__ cbb,
    const float* __restrict__ xpwf, const float* __restrict__ xpwb,
    const float* __restrict__ dtwf, const float* __restrict__ dtbf,
    const float* __restrict__ dtwb, const float* __restrict__ dtbb,
    const float* __restrict__ alogf, const float* __restrict__ alogb,
    const float* __restrict__ Df,   const float* __restrict__ Db,
    const float* __restrict__ mg,   const float* __restrict__ mb,
    const float* __restrict__ outw, const float* __restrict__ l2w,
    const float* __restrict__ l2b,
    float* __restrict__ out) {
  __shared__ float sA[LSEQ * CH];
  __shared__ float sX[LSEQ * CH];
  __shared__ float sXC[LSEQ * CH];
  __shared__ float sXD[LSEQ * 18];
  __shared__ float sV[CH];

  const int b = blockIdx.x;
  const int tid = threadIdx.x;
  float* zb = zws + b * LSEQ * CH;
  float* yb = yws + b * LSEQ * CH;

  // veff[d] = 0.5 * sum_k lin2_w[k] * out_proj_w[k][d]  (out_proj collapsed)
  if (tid < CH) {
    float acc = 0.0f;
    for (int k = 0; k < CH; ++k) acc += l2w[k] * outw[k * CH + tid];
    sV[tid] = acc * 0.5f;
  }

  // Phase 1: comb -> lin1 -> layernorm -> h  (one thread per l)
  if (tid < LSEQ) {
    const float a = ws[b * LSEQ + tid];
    const float m = ws[1024 + b * LSEQ + tid];
    float pre[CH];
    float mu = 0.0f;
    #pragma unroll
    for (int j = 0; j < CH; ++j) {
      float v = a * w1[2 * j] + m * w1[2 * j + 1] + b1[j];
      pre[j] = v;
      mu += v;
    }
    mu *= (1.0f / CH);
    float var = 0.0f;
    #pragma unroll
    for (int j = 0; j < CH; ++j) { float d = pre[j] - mu; var += d * d; }
    var *= (1.0f / CH);
    const float rs = rsqrtf(var + 1e-5f);
    #pragma unroll
    for (int j = 0; j < CH; ++j)
      sA[tid * CH + j] = (pre[j] - mu) * rs * g1[j] + bb1[j];
  }
  __syncthreads();

  // Phase 2: xz = h @ in_proj_w.T  via V_WMMA_F32_16X16X4_F32.
  // Wave w owns M-tile w (rows l = 16w..16w+15); loops over 4 N-tiles (k cols).
  // A layout (ISA 7.12.2, f32 16x4): VGPR r, lanes<16: K=4kk+r; lanes>=16: K=4kk+r+2.
  // B layout assumed mirrored: VGPR r, lanes<16: K row 4kk+r; lanes>=16: 4kk+r+2.
  {
    const int w = tid >> 5;             // wave id == M tile
    const int ln = tid & 31;
    const int m = ln & 15;              // A row / B,C column within tile
    const int koff = (ln >> 4) << 1;    // 0 or 2
    for (int nt = 0; nt < 4; ++nt) {
      v8f acc = {0, 0, 0, 0, 0, 0, 0, 0};
      #pragma unroll
      for (int kk = 0; kk < 8; ++kk) {
        const int kb = kk * 4 + koff;
        v2f av = { sA[(w * 16 + m) * CH + kb],  sA[(w * 16 + m) * CH + kb + 1] };
        v2f bv = { win[(nt * 16 + m) * CH + kb], win[(nt * 16 + m) * CH + kb + 1] };
        acc = __builtin_amdgcn_wmma_f32_16x16x4_f32(
            false, av, false, bv, (short)0, acc, false, false);
      }
      // C/D layout: lanes<16 rows M=r, lanes>=16 rows M=r+8, col = lane&15
      const int row0 = w * 16 + ((ln >> 4) << 3);
      const int col = nt * 16 + m;
      #pragma unroll
      for (int r = 0; r < 8; ++r) {
        const float v = acc[r];
        const int l = row0 + r;
        if (col < CH) sX[l * CH + col] = v;           // x half
        else          zb[l * CH + (col - CH)] = v;    // z half -> workspace
      }
    }
  }
  __syncthreads();

  // Phase 3: the two mamba branches (forward, backward)
  for (int br = 0; br < 2; ++br) {
    const float* cw     = br ? cwb  : cwf;
    const float* cb     = br ? cbb  : cbf;
    const float* xpw    = br ? xpwb : xpwf;
    const float* dtw    = br ? dtwb : dtwf;
    const float* dtbias = br ? dtbb : dtbf;
    const float* alog   = br ? alogb : alogf;
    const float* Dv     = br ? Db   : Df;

    // causal depthwise conv (branch-local time) + silu -> sXC
    for (int idx = tid; idx < LSEQ * CH; idx += 256) {
      const int l = idx >> 5, d = idx & 31;
      float acc = cb[d];
      #pragma unroll
      for (int t = 0; t < 4; ++t) {
        const int ll = l - 3 + t;
        if (ll >= 0) {
          const int src = br ? (LSEQ - 1 - ll) : ll;
          acc += sX[src * CH + d] * cw[d * 4 + t];
        }
      }
      sXC[idx] = silu_f(acc);
    }
    __syncthreads();

    // xdbl = xc @ x_proj_w.T   [128 x 18]
    for (int idx = tid; idx < LSEQ * 18; idx += 256) {
      const int l = idx / 18, r = idx - l * 18;
      float acc = 0.0f;
      #pragma unroll
      for (int d = 0; d < CH; ++d) acc += sXC[l * CH + d] * xpw[r * CH + d];
      sXD[idx] = acc;
    }
    __syncthreads();

    // dt = softplus(dt_raw @ dt_proj_w.T + bias) -> sA (h is dead now)
    for (int idx = tid; idx < LSEQ * CH; idx += 256) {
      const int l = idx >> 5, d = idx & 31;
      const float v =
          sXD[l * 18] * dtw[2 * d] + sXD[l * 18 + 1] * dtw[2 * d + 1] + dtbias[d];
      sA[idx] = (v > 20.0f) ? v : log1pf(expf(v));
    }
    __syncthreads();

    // selective scan: thread (d,n) = (tid>>3, tid&7) owns one state element.
    {
      const int d = tid >> 3, n = tid & 7;
      const float Adn = -expf(alog[d * NST + n]);
      const float Dd = Dv[d];
      float hst = 0.0f;
      for (int l = 0; l < LSEQ; ++l) {
        const float dtl = sA[l * CH + d];
        const float xcl = sXC[l * CH + d];
        const float Bl = sXD[l * 18 + 2 + n];
        const float Cl = sXD[l * 18 + 10 + n];
        hst = expf(dtl * Adn) * hst + dtl * xcl * Bl;
        float val = hst * Cl;
        val += __shfl_xor(val, 1, 32);   // reduce over n within 8-lane group
        val += __shfl_xor(val, 2, 32);
        val += __shfl_xor(val, 4, 32);
        if (n == 0) {
          const int lo = br ? (LSEQ - 1 - l) : l;
          const float zg = zb[lo * CH + d];
          const float yv = (val + xcl * Dd) * silu_f(zg);
          if (br == 0) yb[lo * CH + d] = yv;
          else         yb[lo * CH + d] += yv;
        }
      }
    }
    __syncthreads();
  }

  // Phase 4: layernorm(yf+yb) -> dot(veff) -> tanh -> scale
  if (tid < LSEQ) {
    float v[CH];
    float mu = 0.0f;
    #pragma unroll
    for (int j = 0; j < CH; ++j) { v[j] = yb[tid * CH + j]; mu += v[j]; }
    mu *= (1.0f / CH);
    float var = 0.0f;
    #pragma unroll
    for (int j = 0; j < CH; ++j) { float dd = v[j] - mu; var += dd * dd; }
    var *= (1.0f / CH);
    const float rs = rsqrtf(var + 1e-5f);
    float acc = 0.0f;
    #pragma unroll
    for (int j = 0; j < CH; ++j) {
      const float yn = (v[j] - mu) * rs * mg[j] + mb[j];
      acc += yn * sV[j];
    }
    out[b * LSEQ + tid] = tanhf(acc + l2b[0]) * 0.5f + 1.0f;
  }
}

extern "C" void kernel_launch(void* const* d_in, const int* in_sizes, int n_in,
                              void* d_out, int out_size, void* d_ws, size_t ws_size,
                              hipStream_t stream) {
  (void)in_sizes; (void)n_in; (void)out_size; (void)ws_size;
  const float* input = (const float*)d_in[0];
  float* ws = (float*)d_ws;              // [0..2047]: avg|max
  float* zws = ws + 2048;                // 8*128*32
  float* yws = zws + 8 * 128 * 32;       // 8*128*32

  k_reduce<<<dim3(1024), dim3(1024), 0, stream>>>(input, ws);

  k_mamba<<<dim3(8), dim3(256), 0, stream>>>(
      ws, zws, yws,
      (const float*)d_in[1],  (const float*)d_in[2],   // lin1_w, lin1_b
      (const float*)d_in[3],  (const float*)d_in[4],   // ln1_g, ln1_b
      (const float*)d_in[5],                           // in_proj_w
      (const float*)d_in[6],  (const float*)d_in[7],   // conv_w, conv_b
      (const float*)d_in[8],  (const float*)d_in[9],   // conv_b_w, conv_b_b
      (const float*)d_in[10], (const float*)d_in[11],  // x_proj_w, x_proj_b_w
      (const float*)d_in[12], (const float*)d_in[13],  // dt_proj_w, dt_proj_bias
      (const float*)d_in[14], (const float*)d_in[15],  // dt_proj_b_w, dt_proj_b_bias
      (const float*)d_in[16], (const float*)d_in[17],  // A_log, A_b_log
      (const float*)d_in[18], (const float*)d_in[19],  // Dp, D_b
      (const float*)d_in[20], (const float*)d_in[21],  // mnorm_g, mnorm_b
      (const float*)d_in[22], (const float*)d_in[23],  // out_proj_w, lin2_w
      (const float*)d_in[24],                          // lin2_b
      (float*)d_out);
}